// RerankLoss_52312701665763
// MI455X (gfx1250) — compile-verified
//
#include <hip/hip_runtime.h>

typedef __attribute__((ext_vector_type(2))) float v2f;
typedef __attribute__((ext_vector_type(8))) float v8f;

#define L_LEN 512
#define TILE  16
#define BLOCK 256
#define WAVES (BLOCK / 32)     // 8 wave32s per block

__device__ __forceinline__ float relu1(float v) {
    // v_med3_num_f32(v, 0, FLT_MAX): single-op relu (finite bound avoids the
    // med3(x,0,+inf) -> canonicalize+max InstCombine fold).
    return __builtin_amdgcn_fmed3f(v, 0.0f, __FLT_MAX__);
}

// One block per sample b. Positives/negatives are compacted via a block-wide
// stable prefix sum, so the hinge tile sweep only covers n_pos x n_neg
// (~4x fewer tiles than the full 512x512 masked sweep).
// Each 16x16 tile of the pre-relu matrix is one V_WMMA_F32_16X16X4_F32
// (duplicated rank-2 product; both lane halves identical -> layout-agnostic):
//   A row i (K0..K3) = [ 1, -x_i, 1, -x_i ]          (compacted positives)
//   B col j (K0..K3) = [ 1+x_j, 1, 1+x_j, 1 ]        (compacted negatives)
//   D[i][j] = 2 * (1 - x_i + x_j);  relu; sum; /2 folded into normalization.
// Zero-padded rows/cols contribute exactly zero.
__global__ __launch_bounds__(BLOCK) void rerank_pair_hinge_kernel(
    const float* __restrict__ x, const int* __restrict__ lab,
    float* __restrict__ per_sample)
{
    const int b   = blockIdx.x;
    const int tid = threadIdx.x;

    __shared__ float s_a0[L_LEN];   // A K0/K2 :  1          (compacted pos)
    __shared__ float s_a1[L_LEN];   // A K1/K3 : -x_i
    __shared__ float s_b0[L_LEN];   // B K0/K2 :  1 + x_j    (compacted neg)
    __shared__ float s_b1[L_LEN];   // B K1/K3 :  1
    __shared__ int   s_scan[BLOCK];
    __shared__ float s_red[BLOCK];

    // ---- load this thread's 2 elements -------------------------------------
    const float2 xv2 = ((const float2*)(x + (size_t)b * L_LEN))[tid];
    const int2   lv2 = ((const int2*)(lab + (size_t)b * L_LEN))[tid];
    const int lp = lv2.x + lv2.y;            // positives in this pair (0..2)

    // ---- stable block-wide inclusive scan of positive counts ---------------
    s_scan[tid] = lp;
    __syncthreads();
    #pragma unroll
    for (int off = 1; off < BLOCK; off <<= 1) {
        int v = (tid >= off) ? s_scan[tid - off] : 0;
        __syncthreads();
        s_scan[tid] += v;
        __syncthreads();
    }
    const int excl = s_scan[tid] - lp;       // positives before element 2*tid
    const int np   = s_scan[BLOCK - 1];      // total positives (uniform)

    // ---- scatter compacted WMMA operand factors ----------------------------
    {
        const int k = 2 * tid;
        if (lv2.x) { s_a0[excl] = 1.0f;       s_a1[excl] = -xv2.x; }
        else       { const int nj = k - excl;
                     s_b0[nj] = 1.0f + xv2.x; s_b1[nj] = 1.0f; }
        const int excl1 = excl + lv2.x;
        if (lv2.y) { s_a0[excl1] = 1.0f;       s_a1[excl1] = -xv2.y; }
        else       { const int nj = (k + 1) - excl1;
                     s_b0[nj] = 1.0f + xv2.y;  s_b1[nj] = 1.0f; }
    }

    // ---- zero-pad both compacted arrays to a multiple of TILE --------------
    const int nn      = L_LEN - np;
    const int np_pad  = (np + TILE - 1) & ~(TILE - 1);
    const int nn_pad  = (nn + TILE - 1) & ~(TILE - 1);
    if (tid < TILE) {
        const int i = np + tid;
        if (i < np_pad) { s_a0[i] = 0.0f; s_a1[i] = 0.0f; }
    } else if (tid < 2 * TILE) {
        const int j = nn + (tid - TILE);
        if (j < nn_pad) { s_b0[j] = 0.0f; s_b1[j] = 0.0f; }
    }
    __syncthreads();

    // ---- tile sweep: nt_i x nt_j WMMAs, scalar-uniform loop bounds ---------
    const int wave = __builtin_amdgcn_readfirstlane(tid >> 5);
    const int nt_i = __builtin_amdgcn_readfirstlane(np_pad >> 4);
    const int nt_j = __builtin_amdgcn_readfirstlane(nn_pad >> 4);
    const int l16  = tid & 15;               // lanes t and t+16 index identically

    v8f acc = {0.f, 0.f, 0.f, 0.f, 0.f, 0.f, 0.f, 0.f};

    #pragma unroll 1
    for (int it = wave; it < nt_i; it += WAVES) {
        const int irow = it * TILE + l16;
        v2f a;
        a.x = s_a0[irow];
        a.y = s_a1[irow];

        #pragma unroll 4                     // 4 independent WMMA->relu chains
        for (int jt = 0; jt < nt_j; ++jt) {
            const int jcol = jt * TILE + l16;
            v2f bm;
            bm.x = s_b0[jcol];               // fuses: ds_load_2addr_stride64
            bm.y = s_b1[jcol];

            v8f c = {0.f, 0.f, 0.f, 0.f, 0.f, 0.f, 0.f, 0.f};
            v8f d = __builtin_amdgcn_wmma_f32_16x16x4_f32(
                        false, a, false, bm, (short)0, c, false, false);

            v8f r;
            #pragma unroll
            for (int e = 0; e < 8; ++e) r[e] = relu1(d[e]);
            #pragma unroll
            for (int e = 0; e < 8; ++e) acc[e] += r[e];   // -> v_pk_add_f32
        }
    }

    // ---- deterministic block reduction -------------------------------------
    float s = ((acc[0] + acc[1]) + (acc[2] + acc[3])) +
              ((acc[4] + acc[5]) + (acc[6] + acc[7]));
    s_red[tid] = s;
    __syncthreads();

    #pragma unroll
    for (int st = BLOCK / 2; st > 0; st >>= 1) {
        if (tid < st) s_red[tid] += s_red[tid + st];
        __syncthreads();
    }

    if (tid == 0) {
        const float npairs = (float)np * (float)nn;
        // /2 removes the K-duplication factor (exact: power of two)
        per_sample[b] = s_red[0] / (2.0f * fmaxf(npairs, 1.0f));
    }
}

__global__ __launch_bounds__(BLOCK) void rerank_reduce_kernel(
    const float* __restrict__ per_sample, float* __restrict__ out, int B)
{
    __shared__ float s_red[BLOCK];
    const int tid = threadIdx.x;

    float s = 0.0f;
    for (int i = tid; i < B; i += BLOCK) s += per_sample[i];  // fixed order
    s_red[tid] = s;
    __syncthreads();

    #pragma unroll
    for (int st = BLOCK / 2; st > 0; st >>= 1) {
        if (tid < st) s_red[tid] += s_red[tid + st];
        __syncthreads();
    }

    if (tid == 0) out[0] = s_red[0] / (float)B;
}

extern "C" void kernel_launch(void* const* d_in, const int* in_sizes, int n_in,
                              void* d_out, int out_size, void* d_ws, size_t ws_size,
                              hipStream_t stream)
{
    const float* x   = (const float*)d_in[0];   // [B, 512] fp32 scores
    const int*   lab = (const int*)d_in[1];     // [B, 512] int32 labels
    const int    B   = in_sizes[0] / L_LEN;     // 1024

    float* per_sample = (float*)d_ws;           // B floats of scratch

    rerank_pair_hinge_kernel<<<B, BLOCK, 0, stream>>>(x, lab, per_sample);
    rerank_reduce_kernel<<<1, BLOCK, 0, stream>>>(per_sample, (float*)d_out, B);
}